// AEULoss_27676769255513
// MI455X (gfx1250) — compile-verified
//
#include <hip/hip_runtime.h>

typedef __attribute__((ext_vector_type(2))) float v2f;
typedef __attribute__((ext_vector_type(4))) float v4f;
typedef __attribute__((ext_vector_type(8))) float v8f;

#define NF 4
#define NB 4096
#define ND 512
#define NG (NB / 2)            // 2048 groups per feature
#define NPAIRS (NF * NG)       // 8192 (f,g) wave tasks
#define N_IMG (4096 * 4096)
#define N4_IMG (N_IMG / 4)     // 4,194,304 float4 slots
#define REC_BLOCKS 4096
#define THREADS 256

// Non-temporal 16B load: data is streamed exactly once (working set
// 320 MB > 192 MB L2), so TH=NT is the right CDNA5 cache policy.
// Must use a clang ext_vector type, not HIP_vector_type.
__device__ __forceinline__ v4f ldg_nt(const v4f* p) {
    return __builtin_nontemporal_load(p);
}

// Exact-f32 wave32 reduction through the matrix pipe:
// A(16x4) holds {a0,a1} per lane; B = ones(4x16) => D[m][*] = row-sum of A.
// Summing the 8 D VGPRs + one shfl_xor(16) yields the full 32-lane sum.
__device__ __forceinline__ float wave_reduce_wmma(float a0, float a1) {
    v2f A; A.x = a0;   A.y = a1;
    v2f B; B.x = 1.0f; B.y = 1.0f;
    v8f C = {};
    v8f D = __builtin_amdgcn_wmma_f32_16x16x4_f32(
        /*neg_a=*/false, A, /*neg_b=*/false, B,
        /*c_mod=*/(short)0, C, /*reuse_a=*/false, /*reuse_b=*/false);
    float h = ((D[0] + D[1]) + (D[2] + D[3])) + ((D[4] + D[5]) + (D[6] + D[7]));
    h += __shfl_xor(h, 16, 32);
    return h; // every lane holds the wave total
}

// ---------------- Kernel 1: reconstruction-loss partial sums ----------------
__global__ void rec_partial_kernel(const float* __restrict__ image,
                                   const float* __restrict__ x_recons,
                                   float* __restrict__ block_out) {
    const v4f* img4 = (const v4f*)image;
    const v4f* xr4  = (const v4f*)x_recons;

    float acc0 = 0.0f, acc1 = 0.0f;
    unsigned stride = gridDim.x * blockDim.x;
    for (unsigned i = blockIdx.x * blockDim.x + threadIdx.x; i < N4_IMG; i += stride) {
        v4f v = ldg_nt(img4 + i);           // image read once per slot
        #pragma unroll
        for (int f = 0; f < NF; ++f) {
            v4f r = ldg_nt(xr4 + (size_t)f * N4_IMG + i);
            float dx = v.x - r.x, dy = v.y - r.y;
            float dz = v.z - r.z, dw = v.w - r.w;
            acc0 = fmaf(dx, dx, fmaf(dy, dy, acc0));
            acc1 = fmaf(dz, dz, fmaf(dw, dw, acc1));
        }
    }

    float wsum = wave_reduce_wmma(acc0, acc1);   // full EXEC: all lanes active

    __shared__ float red[THREADS / 32];
    int wid = threadIdx.x >> 5, lane = threadIdx.x & 31;
    if (lane == 0) red[wid] = wsum;
    __syncthreads();
    if (threadIdx.x == 0) {
        float t = 0.0f;
        #pragma unroll
        for (int i = 0; i < THREADS / 32; ++i) t += red[i];
        block_out[blockIdx.x] = t;
    }
}

// ------------- Kernel 2: s[f,g] = 0.25 * ||x_{2g} - x_{2g+1}||^2 ------------
__global__ void sim_pairdist_kernel(const float* __restrict__ features,
                                    float* __restrict__ s_out) {
    unsigned pair = (blockIdx.x * blockDim.x + threadIdx.x) >> 5;  // one wave / (f,g)
    int lane = threadIdx.x & 31;
    unsigned f = pair >> 11;         // pair / 2048
    unsigned g = pair & (NG - 1);

    const v4f* rowA = (const v4f*)(features + ((size_t)f * NB + 2u * g) * ND);
    const v4f* rowB = rowA + (ND / 4);

    float acc0 = 0.0f, acc1 = 0.0f;
    #pragma unroll
    for (int j = 0; j < 4; ++j) {    // 128 float4 slots, lane-strided
        int k = lane + 32 * j;
        v4f a = ldg_nt(rowA + k);
        v4f b = ldg_nt(rowB + k);
        float dx = a.x - b.x, dy = a.y - b.y;
        float dz = a.z - b.z, dw = a.w - b.w;
        if (j & 1) {
            acc1 = fmaf(dx, dx, fmaf(dy, dy, acc1));
            acc1 = fmaf(dz, dz, fmaf(dw, dw, acc1));
        } else {
            acc0 = fmaf(dx, dx, fmaf(dy, dy, acc0));
            acc0 = fmaf(dz, dz, fmaf(dw, dw, acc0));
        }
    }

    float d2 = wave_reduce_wmma(acc0, acc1);
    if (lane == 0) s_out[pair] = 0.25f * d2;
}

// ---------------- Kernel 3: final combine (single block, fixed order) -------
__global__ void finalize_kernel(const float* __restrict__ rec_partials,
                                const float* __restrict__ s_vals,
                                float* __restrict__ out) {
    float r0 = 0.0f, r1 = 0.0f;
    for (int i = threadIdx.x; i < REC_BLOCKS; i += THREADS) {
        if (((i / THREADS) & 1) == 0) r0 += rec_partials[i];
        else                          r1 += rec_partials[i];
    }

    float s0a = 0.0f, s1a = 0.0f;
    for (int g = threadIdx.x; g < NG; g += THREADS) {
        float s[NF];
        #pragma unroll
        for (int f = 0; f < NF; ++f) s[f] = s_vals[f * NG + g];
        float t = 0.0f;
        #pragma unroll
        for (int fi = 0; fi < NF; ++fi)
            #pragma unroll
            for (int fj = fi + 1; fj < NF; ++fj) {
                float num = 4.0f * s[fi] * s[fj];   // cross[f,h,g]
                t += num / fmaxf(num, 1e-8f);       // / max(std_f*std_h, eps)
            }
        if (g & 1) s1a += t; else s0a += t;
    }

    float rec_sum = wave_reduce_wmma(r0, r1);
    float sim_sum = wave_reduce_wmma(s0a, s1a);

    __shared__ float redR[THREADS / 32];
    __shared__ float redS[THREADS / 32];
    int wid = threadIdx.x >> 5, lane = threadIdx.x & 31;
    if (lane == 0) { redR[wid] = rec_sum; redS[wid] = sim_sum; }
    __syncthreads();
    if (threadIdx.x == 0) {
        float R = 0.0f, S = 0.0f;
        #pragma unroll
        for (int i = 0; i < THREADS / 32; ++i) { R += redR[i]; S += redS[i]; }
        R *= 1.0f / 4096.0f;   // mean over last dim (size 4096)
        out[0] = S;
        out[1] = R;
        out[2] = S + R;
    }
}

extern "C" void kernel_launch(void* const* d_in, const int* in_sizes, int n_in,
                              void* d_out, int out_size, void* d_ws, size_t ws_size,
                              hipStream_t stream) {
    const float* x_recons = (const float*)d_in[0];
    const float* features = (const float*)d_in[1];
    const float* image    = (const float*)d_in[2];
    (void)in_sizes; (void)n_in; (void)ws_size; (void)out_size;
    // d_in[3] = log_vars: dead in the reference output (alpha=beta=1).

    float* ws_rec = (float*)d_ws;            // [REC_BLOCKS] floats
    float* ws_s   = ws_rec + REC_BLOCKS;     // [NF * NG] floats
    float* out    = (float*)d_out;

    rec_partial_kernel<<<REC_BLOCKS, THREADS, 0, stream>>>(image, x_recons, ws_rec);
    sim_pairdist_kernel<<<NPAIRS / (THREADS / 32), THREADS, 0, stream>>>(features, ws_s);
    finalize_kernel<<<1, THREADS, 0, stream>>>(ws_rec, ws_s, out);
}